// LearnableTD_36756330119251
// MI455X (gfx1250) — compile-verified
//
#include <hip/hip_runtime.h>

// LearnableTD lambda-return scan for MI455X (gfx1250, wave32).
//  - 1 thread per batch row, reverse scan over S in float4 chunks.
//  - lambda table + gamma precomputed once to workspace; lambda staged into
//    LDS via CDNA5 async global->LDS (ASYNCcnt path, global_load_async_to_lds_b128).
//  - non-temporal b128 stores for the 256MB of outputs (protect 192MB L2).

#define B_SZ 32768
#define S_SZ 1024
#define DF_C 0.99f
#define AL_C 0.95f
#define BLK 128

typedef float v4f __attribute__((ext_vector_type(4)));

// ws layout: ws[0..S-1] = lambda table, ws[S] = gamma.
__global__ void td_precompute_kernel(const float* __restrict__ raw_lambd,
                                     const float* __restrict__ raw_gamma,
                                     float* __restrict__ ws) {
  int t = blockIdx.x * blockDim.x + threadIdx.x;
  if (t < S_SZ) {
    float x = raw_lambd[t];
    float s = 1.0f / (1.0f + __expf(-x));
    ws[t] = AL_C + (1.0f - AL_C) * (2.0f * s - 1.0f);
  }
  if (t == 0) {
    float x = raw_gamma[0];
    float s = 1.0f / (1.0f + __expf(-x));
    ws[S_SZ] = DF_C + (1.0f - DF_C) * (2.0f * s - 1.0f);
  }
}

__global__ __launch_bounds__(BLK) void td_scan_kernel(
    const float* __restrict__ values,  // (B, S+1)
    const float* __restrict__ rewards, // (B, S)
    const float* __restrict__ dones,   // (B, S)
    const float* __restrict__ ws,      // [S] lambda table, [S] = gamma
    float* __restrict__ out) {         // (2*B*S): [lambda_returns | sum_rewards]
  __shared__ float s_lam[S_SZ];
  const int tid = threadIdx.x;

  // ---- CDNA5 async global->LDS staging of the 4KB lambda table ----
  // Each lane moves 16B per op; completion tracked with ASYNCcnt.
  for (int i = tid * 4; i < S_SZ; i += BLK * 4) {
    unsigned lds_off = (unsigned)(size_t)(&s_lam[i]); // low 32 bits = LDS offset
    const float* gp = ws + i;
    asm volatile("global_load_async_to_lds_b128 %0, %1, off"
                 :
                 : "v"(lds_off), "v"(gp)
                 : "memory");
  }

  const float gamma = ws[S_SZ]; // uniform scalar, L2-hot

  const int b = blockIdx.x * BLK + tid;
  const float* vrow = values + (size_t)b * (S_SZ + 1);
  const float* rrow = rewards + (size_t)b * S_SZ;
  const float* drow = dones + (size_t)b * S_SZ;
  float* lr_out = out + (size_t)b * S_SZ;
  float* sr_out = out + (size_t)B_SZ * S_SZ + (size_t)b * S_SZ;

  float sr = 0.0f;
  float lr = vrow[S_SZ]; // bootstrap with values[:, -1]

  asm volatile("s_wait_asynccnt 0" ::: "memory");
  __syncthreads();

  // Reverse scan, 4 timesteps per chunk (b128 loads/stores).
  for (int tc = S_SZ - 4; tc >= 0; tc -= 4) {
    v4f r4 = *(const v4f*)(rrow + tc);
    v4f d4 = *(const v4f*)(drow + tc);
    float v0 = vrow[tc + 1];
    float v1 = vrow[tc + 2];
    float v2 = vrow[tc + 3];
    float v3 = vrow[tc + 4];
    v4f l4 = *(const v4f*)(&s_lam[tc]); // ds_load_b128

    v4f lr4, sr4;
    {
      float g = gamma * (1.0f - d4.w);
      sr = r4.w + g * (l4.w * sr);
      lr = r4.w + g * ((1.0f - l4.w) * v3 + l4.w * lr);
      lr4.w = lr; sr4.w = sr;
    }
    {
      float g = gamma * (1.0f - d4.z);
      sr = r4.z + g * (l4.z * sr);
      lr = r4.z + g * ((1.0f - l4.z) * v2 + l4.z * lr);
      lr4.z = lr; sr4.z = sr;
    }
    {
      float g = gamma * (1.0f - d4.y);
      sr = r4.y + g * (l4.y * sr);
      lr = r4.y + g * ((1.0f - l4.y) * v1 + l4.y * lr);
      lr4.y = lr; sr4.y = sr;
    }
    {
      float g = gamma * (1.0f - d4.x);
      sr = r4.x + g * (l4.x * sr);
      lr = r4.x + g * ((1.0f - l4.x) * v0 + l4.x * lr);
      lr4.x = lr; sr4.x = sr;
    }

    // Streaming outputs: bypass/deprioritize caches (written once, never read).
    __builtin_nontemporal_store(lr4, (v4f*)(lr_out + tc));
    __builtin_nontemporal_store(sr4, (v4f*)(sr_out + tc));
  }
}

extern "C" void kernel_launch(void* const* d_in, const int* in_sizes, int n_in,
                              void* d_out, int out_size, void* d_ws, size_t ws_size,
                              hipStream_t stream) {
  (void)in_sizes; (void)n_in; (void)out_size; (void)ws_size;
  const float* values    = (const float*)d_in[0]; // (B, S+1, 1)
  const float* rewards   = (const float*)d_in[1]; // (B, S, 1)
  const float* dones     = (const float*)d_in[2]; // (B, S, 1)
  const float* raw_gamma = (const float*)d_in[3]; // (1,)
  const float* raw_lambd = (const float*)d_in[4]; // (S,)
  float* out = (float*)d_out;
  float* ws  = (float*)d_ws; // S+1 floats scratch: lambda table + gamma

  td_precompute_kernel<<<(S_SZ + 255) / 256, 256, 0, stream>>>(raw_lambd,
                                                               raw_gamma, ws);
  td_scan_kernel<<<B_SZ / BLK, BLK, 0, stream>>>(values, rewards, dones, ws, out);
}